// Conv2d_67233418051840
// MI455X (gfx1250) — compile-verified
//
#include <hip/hip_runtime.h>

typedef __attribute__((ext_vector_type(2))) float v2f;
typedef __attribute__((ext_vector_type(8))) float v8f;

#define NB   16
#define CIN  128
#define H    112
#define W    112
#define KOUT 256
#define HO   110
#define WO   110
#define LDSP 132          // padded LDS row pitch in floats (rows = w-position, cols = channel)
#define WREP_ELEMS (9 * KOUT * CIN)   // repacked weights: [tap][m][c]

// One-time weight repack: wgt[m][c][r][s] -> wrep[(tap*KOUT + m)*CIN + c]
__global__ __launch_bounds__(256)
void repack_weights(const float* __restrict__ wgt, float* __restrict__ wrep)
{
    int o   = blockIdx.x * 256 + threadIdx.x;   // 0 .. 294911
    int c   = o & (CIN - 1);
    int m   = (o >> 7) & (KOUT - 1);
    int tap = o >> 15;
    wrep[o] = wgt[(m * CIN + c) * 9 + tap];
}

template <bool REPACK>
__global__ __launch_bounds__(256)
void conv3x3_wmma_f32(const float* __restrict__ x,
                      const float* __restrict__ wgt,
                      const float* __restrict__ wrep,
                      const float* __restrict__ bias,
                      float* __restrict__ out)
{
    // Transposed tile: ldsB[j * LDSP + c], j = w-position 0..65, c = channel 0..127.
    __shared__ float ldsB[66 * LDSP];   // 34,848 B

    const int bid = blockIdx.x;
    const int mb  = bid & 1;            // which 128 output channels
    const int seg = (bid >> 1) & 1;     // ow segment (0 or 46)
    const int oh  = (bid >> 2) % HO;    // 0..109
    const int b   = bid / (4 * HO);     // 0..15

    const int tid  = threadIdx.x;
    const int lane = tid & 31;
    const int wave = tid >> 5;          // 0..7

    const int ow0   = seg * 46;         // segments [0,63] & [46,109]; overlap writes identical values
    const int mBase = mb * 128 + wave * 16;
    const int mrow  = lane & 15;        // A-matrix row owned by this lane
    const int kk    = (lane >> 4) << 1; // K sub-offset: 0 (lanes 0-15) or 2 (lanes 16-31)
    const int ncol  = lane & 15;        // B/D column owned by this lane

    v8f zero = {0.f,0.f,0.f,0.f,0.f,0.f,0.f,0.f};
    v8f acc[4];
    acc[0] = zero; acc[1] = zero; acc[2] = zero; acc[3] = zero;

    // Per-thread staging coordinates (shift/mask only, coalesced in e):
    //   main: 128x64 elements, e = i*256+tid, c = e>>6, j = e&63
    //   tail: columns j=64,65 for all c: c = tid>>1, j = 64+(tid&1)
    const int tc = tid >> 1;
    const int tj = 64 + (tid & 1);

    for (int r = 0; r < 3; ++r) {
        const float* src = x + (((long)b * CIN) * H + (oh + r)) * W + ow0;
        if (r) __syncthreads();
        #pragma unroll
        for (int i = 0; i < 32; ++i) {           // 128*64 / 256 = 32
            int e = i * 256 + tid;
            int c = e >> 6;
            int j = e & 63;
            ldsB[j * LDSP + c] = src[(long)c * (H * W) + j];
        }
        ldsB[tj * LDSP + tc] = src[(long)tc * (H * W) + tj];   // halo tail
        if (r < 2)                                             // prefetch next x row
            __builtin_prefetch(src + W + (long)(tid >> 1) * (H * W), 0, 1);
        __syncthreads();

        for (int s = 0; s < 3; ++s) {
            const int tap = r * 3 + s;
            // A base: repacked [tap][m][c] -> contiguous b64 per K-step; else strided gather.
            const float* wA = REPACK
                ? wrep + ((long)(tap * KOUT + mBase + mrow)) * CIN + kk
                : wgt  + ((long)(mBase + mrow) * CIN + kk) * 9 + tap;
            const float* bp = &ldsB[(s + ncol) * LDSP + kk];
            #pragma unroll 4
            for (int c0 = 0; c0 < CIN; c0 += 4) {
                v2f A;
                if (REPACK) {
                    A = *(const v2f*)(wA + c0);              // one global_load_b64
                } else {
                    A.x = wA[(c0    ) * 9];
                    A.y = wA[(c0 + 1) * 9];
                }
                #pragma unroll
                for (int t = 0; t < 4; ++t) {                // 4 n-subtiles reuse A regs
                    // one ds_load_b64 straight into the operand pair
                    v2f B = *(const v2f*)(bp + t * 16 * LDSP + c0);
                    acc[t] = __builtin_amdgcn_wmma_f32_16x16x4_f32(
                        false, A, false, B, (short)0, acc[t], false, false);
                }
            }
        }
    }

    // Epilogue: bias + store. C/D layout: VGPR v, lane-half h -> M = v + 8*h, N = lane&15.
    const int mhalf = (lane >> 4) << 3;  // 0 or 8
    #pragma unroll
    for (int t = 0; t < 4; ++t) {
        int ow = ow0 + t * 16 + ncol;    // <= 109, always in bounds
        #pragma unroll
        for (int v = 0; v < 8; ++v) {
            int m = mBase + mhalf + v;
            out[(((long)b * KOUT + m) * HO + oh) * WO + ow] = acc[t][v] + bias[m];
        }
    }
}

extern "C" void kernel_launch(void* const* d_in, const int* in_sizes, int n_in,
                              void* d_out, int out_size, void* d_ws, size_t ws_size,
                              hipStream_t stream) {
    const float* x    = (const float*)d_in[0];
    const float* wgt  = (const float*)d_in[1];
    const float* bias = (const float*)d_in[2];
    float* out  = (float*)d_out;
    float* wrep = (float*)d_ws;

    dim3 grid(2 /*m*/ * 2 /*seg*/ * HO * NB);   // 7040 blocks
    dim3 block(256);                             // 8 wave32s

    if (ws_size >= (size_t)WREP_ELEMS * sizeof(float)) {
        hipLaunchKernelGGL(repack_weights, dim3(WREP_ELEMS / 256), block, 0, stream, wgt, wrep);
        hipLaunchKernelGGL((conv3x3_wmma_f32<true>), grid, block, 0, stream,
                           x, wgt, wrep, bias, out);
    } else {
        hipLaunchKernelGGL((conv3x3_wmma_f32<false>), grid, block, 0, stream,
                           x, wgt, wgt, bias, out);
    }
}